// PGN_40561671144097
// MI455X (gfx1250) — compile-verified
//
#include <hip/hip_runtime.h>

// Pointer-generator final distribution for MI455X (gfx1250).
//
// out[b,t, :V]    = p_gen[b,t] * vocab[b,t,:]          (scaled generation dist)
// out[b,t, V:Vx]  = 0                                  (in-article OOV pad)
// out[b,t, ids[b,l]] += (1-p_gen[b,t]) * attn[b,t,l]   (scatter-add copy dist)
//
// Strategy: one block per (b,t) row. The whole extended row (Vext=50100 f32 =
// 200.4 KB) lives in LDS (CDNA5 WGP has 320 KB). Scatter uses ds_add_f32 LDS
// atomics (no global atomics), and the row is flushed to HBM with
// global_store_async_from_lds_b128 DMA stores (ASYNCcnt path) — the output
// never round-trips through VGPRs.

__global__ void __launch_bounds__(512)
pgn_final_dist_kernel(const float* __restrict__ vocab,
                      const float* __restrict__ attn,
                      const float* __restrict__ pgen,
                      const int*   __restrict__ ids,
                      float*       __restrict__ out,
                      int T, int L, int V, int Vext)
{
    extern __shared__ float lds[];

    const int bt   = blockIdx.x;          // flattened (b, t)
    const int b    = bt / T;
    const int tid  = threadIdx.x;
    const int nthr = blockDim.x;

    const float pg  = pgen[bt];
    const float cpw = 1.0f - pg;

    // ---- Phase 1: lds[0..V) = pg * vocab_row ; lds[V..Vext) = 0 -------------
    const float4* __restrict__ vrow4 =
        reinterpret_cast<const float4*>(vocab + (size_t)bt * V);
    float4* __restrict__ lds4 = reinterpret_cast<float4*>(lds);
    const int V4 = V >> 2;
    for (int i = tid; i < V4; i += nthr) {
        float4 v = vrow4[i];                      // global_load_b128
        v.x *= pg; v.y *= pg; v.z *= pg; v.w *= pg;
        lds4[i] = v;                              // ds_store_b128
    }
    for (int i = (V4 << 2) + tid; i < V; i += nthr)    // V%4 tail (none here)
        lds[i] = pg * vocab[(size_t)bt * V + i];
    for (int i = V + tid; i < Vext; i += nthr)         // OOV pad region
        lds[i] = 0.0f;
    __syncthreads();

    // ---- Phase 2: scatter-add copy distribution into the LDS row ------------
    const float* __restrict__ arow = attn + (size_t)bt * L;
    const int*   __restrict__ irow = ids  + (size_t)b  * L;
    for (int l = tid; l < L; l += nthr)
        atomicAdd(&lds[irow[l]], cpw * arow[l]);       // ds_add_f32
    __syncthreads();

    // ---- Phase 3: DMA the finished row LDS -> HBM (async b128 stores) -------
    float* __restrict__ orow = out + (size_t)bt * Vext;
    const int Vx4 = Vext >> 2;
    for (int i = tid; i < Vx4; i += nthr) {
        unsigned long long ga =
            (unsigned long long)(const void*)(orow + (i << 2));
        // Flat LDS pointer low 32 bits == LDS byte address (ISA aperture rule).
        unsigned int la =
            (unsigned int)(unsigned long long)(const void*)(lds + (i << 2));
        asm volatile("global_store_async_from_lds_b128 %0, %1, off"
                     :: "v"(ga), "v"(la) : "memory");
    }
    for (int i = (Vx4 << 2) + tid; i < Vext; i += nthr) // Vext%4 tail (none)
        orow[i] = lds[i];
    asm volatile("s_wait_asynccnt 0" ::: "memory");
}

extern "C" void kernel_launch(void* const* d_in, const int* in_sizes, int n_in,
                              void* d_out, int out_size, void* d_ws, size_t ws_size,
                              hipStream_t stream)
{
    const float* vocab = (const float*)d_in[0];   // [B, T, V]
    const float* attn  = (const float*)d_in[1];   // [B, T, L]
    const float* pgen  = (const float*)d_in[2];   // [B, T, 1]
    const int*   ids   = (const int*)  d_in[3];   // [B, L]
    // d_in[4] = batch_oov_len (scalar); Vext is derived from out_size instead
    // so we never need a device read on the host path (graph-capture safe).

    const int BT   = in_sizes[2];            // B*T   (p_gens element count)
    const int L    = in_sizes[1] / BT;       // attn  = B*T*L
    const int B    = in_sizes[3] / L;        // ids   = B*L
    const int T    = BT / B;
    const int V    = in_sizes[0] / BT;       // vocab = B*T*V
    const int Vext = out_size / BT;          // V + batch_oov_len

    const size_t shmem = (size_t)Vext * sizeof(float);   // 200.4 KB < 320 KB
    (void)hipFuncSetAttribute((const void*)pgn_final_dist_kernel,
                              hipFuncAttributeMaxDynamicSharedMemorySize,
                              (int)shmem);

    pgn_final_dist_kernel<<<dim3(BT), dim3(512), shmem, stream>>>(
        vocab, attn, pgen, ids, (float*)d_out, T, L, V, Vext);
}